// multi_head_27848567947950
// MI455X (gfx1250) — compile-verified
//
#include <hip/hip_runtime.h>
#include <hip/hip_bf16.h>
#include <math.h>

// ---------------------------------------------------------------------------
// Problem constants (from reference): B=8, S=1024, E=768, H=12, DK=DV=64
// ---------------------------------------------------------------------------
constexpr int Bb = 8;
constexpr int Ss = 1024;
constexpr int Ee = 768;
constexpr int Hh = 12;
constexpr int Dk = 64;
constexpr int Mrows = Bb * Ss;           // 8192
#define SCALE_C 0.125f                   // 1/sqrt(64)
#define EPS_C 1e-5f

typedef __attribute__((ext_vector_type(2))) float v2f;
typedef __attribute__((ext_vector_type(8))) float v8f;
typedef __attribute__((ext_vector_type(4))) int   v4i;

// D = A(16x4 f32) * B(4x16 f32) + C(16x16 f32), native fp32 WMMA (CDNA5).
__device__ __forceinline__ v8f wmma_f32(v2f a, v2f b, v8f c) {
    return __builtin_amdgcn_wmma_f32_16x16x4_f32(false, a, false, b,
                                                 (short)0, c, false, false);
}

// ---------------------------------------------------------------------------
// Async global -> LDS copy (CDNA5 GLOBAL_LOAD_ASYNC_TO_LDS_B128, ASYNCcnt).
// Builtin signature (from hipcc diagnostic): (v4i AS1*, v4i AS3*, imm, imm).
// Per-lane 16B transfer, LDS written directly (no VGPR round trip).
// Guarded: falls back to a synchronous copy if the builtin is unavailable.
// All LDS destinations below are 16-byte aligned (row strides are multiples
// of 4 floats).
// ---------------------------------------------------------------------------
#if defined(__gfx1250__) && \
    __has_builtin(__builtin_amdgcn_global_load_async_to_lds_b128) && \
    __has_builtin(__builtin_amdgcn_s_wait_asynccnt)
#define HAVE_ASYNC_LDS 1
#else
#define HAVE_ASYNC_LDS 0
#endif

__device__ __forceinline__ void async_copy16(const float* g, float* l) {
#if HAVE_ASYNC_LDS
    __builtin_amdgcn_global_load_async_to_lds_b128(
        (__attribute__((address_space(1))) v4i*)(g),
        (__attribute__((address_space(3))) v4i*)(l), 0, 0);
#else
    *(float4*)l = *(const float4*)g;
#endif
}

__device__ __forceinline__ void wait_async() {
#if HAVE_ASYNC_LDS
    __builtin_amdgcn_s_wait_asynccnt(0);
#endif
}

// ---------------------------------------------------------------------------
// GEMM: C[M,N] = A[M,768] @ W[768,N] + bias[N] (+ resid[M,N] optionally)
// Block: 256 threads = 8 waves. Block tile 128x64; each wave 16 rows x 64
// cols (four 16x16 fp32 WMMA accumulators -> 4 WMMAs per A fragment).
// K staged 16/iter into double-buffered LDS via async-to-LDS copies; next
// tile's transfer overlaps current tile's WMMAs.
// LDS pads: As stride 20 floats (80B, 16B-aligned; column reads hit banks
// 20*ln mod 64, all distinct). Bs stride 72 floats (288B; half-waves offset
// by 2*72 = +16 banks -> disjoint from the 16-lane row reads).
// ---------------------------------------------------------------------------
__global__ __launch_bounds__(256) void gemm_bias_kernel(
    const float* __restrict__ A, const float* __restrict__ W,
    const float* __restrict__ bias, const float* __restrict__ resid,
    float* __restrict__ C, int N)
{
    __shared__ float As[2][128][20];
    __shared__ float Bs[2][16][72];

    const int tid  = threadIdx.x;
    const int lid  = tid & 31;
    const int w    = tid >> 5;        // wave 0..7 -> M sub-tile
    const int half = lid >> 4;        // 0: lanes 0-15, 1: lanes 16-31
    const int ln   = lid & 15;
    const int ksel = half * 2;        // K-pair select for A/B fragments

    const int m0 = blockIdx.y * 128;
    const int n0 = blockIdx.x * 64;

    auto stage = [&](int buf, int k0) {
        // A tile 128x16: 512 float4 -> 2 per thread
        #pragma unroll
        for (int i = 0; i < 2; ++i) {
            int j   = i * 256 + tid;
            int row = j >> 2;
            int c4  = (j & 3) * 4;
            async_copy16(A + (size_t)(m0 + row) * Ee + k0 + c4,
                         &As[buf][row][c4]);
        }
        // W tile 16x64: 256 float4 -> 1 per thread
        {
            int row = tid >> 4;
            int c4  = (tid & 15) * 4;
            async_copy16(W + (size_t)(k0 + row) * N + n0 + c4,
                         &Bs[buf][row][c4]);
        }
    };

    v8f acc[4] = {};

    stage(0, 0);
    int buf = 0;
    for (int k0 = 0; k0 < Ee; k0 += 16) {
        wait_async();          // this wave's staged tile is in LDS
        __syncthreads();       // all waves' tiles visible; prev compute done
        if (k0 + 16 < Ee) stage(buf ^ 1, k0 + 16);   // overlap next transfer

        #pragma unroll
        for (int kk = 0; kk < 16; kk += 4) {
            v2f a;
            a.x = As[buf][w * 16 + ln][kk + ksel];
            a.y = As[buf][w * 16 + ln][kk + ksel + 1];
            #pragma unroll
            for (int n = 0; n < 4; ++n) {
                v2f b;
                b.x = Bs[buf][kk + ksel][n * 16 + ln];
                b.y = Bs[buf][kk + ksel + 1][n * 16 + ln];
                acc[n] = wmma_f32(a, b, acc[n]);
            }
        }
        buf ^= 1;
    }

    // Epilogue: C/D layout -> VGPR r: row = r (lanes 0-15) or r+8 (lanes 16-31)
    const int mrow = m0 + w * 16;
    #pragma unroll
    for (int n = 0; n < 4; ++n) {
        int col  = n0 + n * 16 + ln;
        float bn = bias[col];
        #pragma unroll
        for (int r = 0; r < 8; ++r) {
            int row    = mrow + r + half * 8;
            size_t off = (size_t)row * N + col;
            float v    = acc[n][r] + bn;
            if (resid) v += resid[off];
            C[off] = v;
        }
    }
}

// ---------------------------------------------------------------------------
// Flash-style attention. Raw reshape => head h owns contiguous rows
// [h*S, h*S+S) of the [B, S*H, 64] Q/K/V arrays.
// Block: 128 threads = 4 waves; each wave owns a 16-row Q tile (block: 64 q
// rows). K/V staged 64x64 in double-buffered LDS via async-to-LDS; scores
// and P@V via fp32 WMMA; online softmax stats per-row in registers
// (half-wave shfl reductions match the C-matrix row layout).
// masked_fill value is 1e-9 (faithful to reference, NOT -inf).
// LDS pads: Ks stride 68 (K^T column reads -> banks 4*ln, halves +2:
// disjoint); Vs stride 72 (row reads, halves +16 banks: disjoint).
// ---------------------------------------------------------------------------
__global__ __launch_bounds__(128) void attn_kernel(
    const float* __restrict__ Qg, const float* __restrict__ Kg,
    const float* __restrict__ Vg, const unsigned char* __restrict__ maskp,
    float* __restrict__ ctx)
{
    __shared__ float Ks[2][64][68];
    __shared__ float Vs[2][64][72];
    __shared__ float Ps[4][16][17];   // per-wave P tile (VALU-written)

    const int tid  = threadIdx.x;
    const int lid  = tid & 31;
    const int w    = tid >> 5;        // wave 0..3
    const int half = lid >> 4;
    const int ln   = lid & 15;
    const int ksel = half * 2;

    const int bh = blockIdx.y;
    const int b  = bh / Hh;
    const int h  = bh % Hh;
    const int qbase = blockIdx.x * 64 + w * 16;

    const size_t headoff = ((size_t)b * (Hh * (size_t)Ss) + (size_t)h * Ss) * Dk;
    const float* Qh = Qg + headoff;
    const float* Kh = Kg + headoff;
    const float* Vh = Vg + headoff;

    auto stage_kv = [&](int buf, int kb) {
        #pragma unroll
        for (int i = 0; i < 8; ++i) {
            int j   = i * 128 + tid;    // float4 index 0..1023
            int row = j >> 4;
            int c4  = (j & 15) * 4;
            async_copy16(Kh + (size_t)(kb + row) * Dk + c4, &Ks[buf][row][c4]);
            async_copy16(Vh + (size_t)(kb + row) * Dk + c4, &Vs[buf][row][c4]);
        }
    };

    // Q fragments for this wave's 16 rows: 16 chunks of K=4 (DK=64).
    v2f qa[16];
    {
        const float* qrow = Qh + (size_t)(qbase + ln) * Dk;
        #pragma unroll
        for (int f = 0; f < 16; ++f) {
            qa[f].x = qrow[4 * f + ksel];
            qa[f].y = qrow[4 * f + ksel + 1];
        }
    }

    float Mr[8], Lr[8];
    #pragma unroll
    for (int r = 0; r < 8; ++r) { Mr[r] = -INFINITY; Lr[r] = 0.0f; }
    v8f o0 = {}, o1 = {}, o2 = {}, o3 = {};

    stage_kv(0, 0);
    int buf = 0;
    for (int kb = 0; kb < Ss; kb += 64) {
        wait_async();
        __syncthreads();
        if (kb + 64 < Ss) stage_kv(buf ^ 1, kb + 64);  // overlap next transfer

        for (int t = 0; t < 4; ++t) {
            const int kt = t * 16;
            // scores tile 16x16 = sum_k Q[16x4] * K^T[4x16]
            v8f s = {};
            #pragma unroll
            for (int f = 0; f < 16; ++f) {
                v2f bb;   // B[k][n] = K[n][k]
                bb.x = Ks[buf][kt + ln][4 * f + ksel];
                bb.y = Ks[buf][kt + ln][4 * f + ksel + 1];
                s = wmma_f32(qa[f], bb, s);
            }
            // mask + scale (masked -> 1e-9, else *SCALE)
            const size_t mbase = (size_t)b * Ss * Ss;
            #pragma unroll
            for (int r = 0; r < 8; ++r) {
                int qrow = qbase + r + half * 8;
                int kcol = kb + kt + ln;
                unsigned char mk = maskp[mbase + (size_t)qrow * Ss + kcol];
                s[r] = mk ? 1e-9f : s[r] * SCALE_C;
            }
            // online softmax update; reductions stay within 16-lane halves
            float pr[8];
            #pragma unroll
            for (int r = 0; r < 8; ++r) {
                float tm = s[r];
                tm = fmaxf(tm, __shfl_xor(tm, 1, 32));
                tm = fmaxf(tm, __shfl_xor(tm, 2, 32));
                tm = fmaxf(tm, __shfl_xor(tm, 4, 32));
                tm = fmaxf(tm, __shfl_xor(tm, 8, 32));
                float newM = fmaxf(Mr[r], tm);
                float sc   = __expf(Mr[r] - newM);
                Mr[r] = newM;
                float p = __expf(s[r] - newM);
                float ts = p;
                ts += __shfl_xor(ts, 1, 32);
                ts += __shfl_xor(ts, 2, 32);
                ts += __shfl_xor(ts, 4, 32);
                ts += __shfl_xor(ts, 8, 32);
                Lr[r] = Lr[r] * sc + ts;
                o0[r] *= sc; o1[r] *= sc; o2[r] *= sc; o3[r] *= sc;
                pr[r] = p;
            }
            // Stage P (C-layout -> row-major LDS) for use as WMMA A operand
            #pragma unroll
            for (int r = 0; r < 8; ++r)
                Ps[w][r + half * 8][ln] = pr[r];
            __syncthreads();  // cross-lane LDS visibility (uniform for block)
            // ctx += P[16x16] @ V[16x64]
            #pragma unroll
            for (int kc = 0; kc < 16; kc += 4) {
                v2f a;
                a.x = Ps[w][ln][kc + ksel];
                a.y = Ps[w][ln][kc + ksel + 1];
                v2f bb;
                bb.x = Vs[buf][kt + kc + ksel][0 + ln];
                bb.y = Vs[buf][kt + kc + ksel + 1][0 + ln];
                o0 = wmma_f32(a, bb, o0);
                bb.x = Vs[buf][kt + kc + ksel][16 + ln];
                bb.y = Vs[buf][kt + kc + ksel + 1][16 + ln];
                o1 = wmma_f32(a, bb, o1);
                bb.x = Vs[buf][kt + kc + ksel][32 + ln];
                bb.y = Vs[buf][kt + kc + ksel + 1][32 + ln];
                o2 = wmma_f32(a, bb, o2);
                bb.x = Vs[buf][kt + kc + ksel][48 + ln];
                bb.y = Vs[buf][kt + kc + ksel + 1][48 + ln];
                o3 = wmma_f32(a, bb, o3);
            }
        }
        buf ^= 1;
    }

    // Normalize and store: ctx layout [B, S, H*DV] (swapaxes+merge of heads)
    float* outb = ctx + (size_t)b * Ss * (Hh * Dk) + (size_t)h * Dk;
    #pragma unroll
    for (int r = 0; r < 8; ++r) {
        float inv = 1.0f / Lr[r];
        int qrow = qbase + r + half * 8;
        float* orow = outb + (size_t)qrow * (Hh * Dk);
        orow[ 0 + ln] = o0[r] * inv;
        orow[16 + ln] = o1[r] * inv;
        orow[32 + ln] = o2[r] * inv;
        orow[48 + ln] = o3[r] * inv;
    }
}

// ---------------------------------------------------------------------------
// Row LayerNorm over E=768. One block (256 threads) per row.
// ---------------------------------------------------------------------------
__global__ __launch_bounds__(256) void layernorm_kernel(
    const float* __restrict__ X, const float* __restrict__ gamma,
    const float* __restrict__ beta, float* __restrict__ out)
{
    __shared__ float red[256];
    __shared__ float red2[256];
    const int row = blockIdx.x;
    const float* x = X + (size_t)row * Ee;
    float v[3];
    float s = 0.0f, s2 = 0.0f;
    #pragma unroll
    for (int i = 0; i < 3; ++i) {
        v[i] = x[threadIdx.x + i * 256];
        s += v[i];
        s2 += v[i] * v[i];
    }
    red[threadIdx.x] = s;
    red2[threadIdx.x] = s2;
    __syncthreads();
    for (int off = 128; off > 0; off >>= 1) {
        if (threadIdx.x < off) {
            red[threadIdx.x]  += red[threadIdx.x + off];
            red2[threadIdx.x] += red2[threadIdx.x + off];
        }
        __syncthreads();
    }
    const float mean = red[0] * (1.0f / Ee);
    const float var  = red2[0] * (1.0f / Ee) - mean * mean;
    const float rstd = rsqrtf(var + EPS_C);
    float* orow = out + (size_t)row * Ee;
    #pragma unroll
    for (int i = 0; i < 3; ++i) {
        int c = threadIdx.x + i * 256;
        orow[c] = gamma[c] * (v[i] - mean) * rstd + beta[c];
    }
}

// ---------------------------------------------------------------------------
// Launch: QKV GEMMs -> attention -> output GEMM(+bias+residual) -> LayerNorm
// Workspace: Q | K | V | ctx  (4 x 8192*768 floats ~ 101 MB); projection
// output reuses the Q slab (Q dead after attention).
// ---------------------------------------------------------------------------
extern "C" void kernel_launch(void* const* d_in, const int* in_sizes, int n_in,
                              void* d_out, int out_size, void* d_ws, size_t ws_size,
                              hipStream_t stream) {
    const float* Input = (const float*)d_in[0];
    const unsigned char* mask = (const unsigned char*)d_in[1];  // jnp bool: 1B
    const float* Wq = (const float*)d_in[2];
    const float* bq = (const float*)d_in[3];
    const float* Wk = (const float*)d_in[4];
    const float* bk = (const float*)d_in[5];
    const float* Wv = (const float*)d_in[6];
    const float* bv = (const float*)d_in[7];
    const float* Wo = (const float*)d_in[8];
    const float* bo = (const float*)d_in[9];
    const float* gamma = (const float*)d_in[10];
    const float* beta  = (const float*)d_in[11];
    float* out = (float*)d_out;

    const size_t slab = (size_t)Mrows * Ee;   // 8192*768 floats
    float* Q   = (float*)d_ws;
    float* K   = Q + slab;
    float* V   = K + slab;
    float* ctx = V + slab;
    float* proj = Q;  // reuse: Q dead after attention

    dim3 gGemm(Ee / 64, Mrows / 128);   // 12 x 64
    gemm_bias_kernel<<<gGemm, 256, 0, stream>>>(Input, Wq, bq, nullptr, Q, Ee);
    gemm_bias_kernel<<<gGemm, 256, 0, stream>>>(Input, Wk, bk, nullptr, K, Ee);
    gemm_bias_kernel<<<gGemm, 256, 0, stream>>>(Input, Wv, bv, nullptr, V, Ee);

    dim3 gAttn(Ss / 64, Bb * Hh);       // 16 x 96
    attn_kernel<<<gAttn, 128, 0, stream>>>(Q, K, V, mask, ctx);

    gemm_bias_kernel<<<gGemm, 256, 0, stream>>>(ctx, Wo, bo, Input, proj, Ee);

    layernorm_kernel<<<Mrows, 256, 0, stream>>>(proj, gamma, beta, out);
}